// PointerGenNetwork_32899449487739
// MI455X (gfx1250) — compile-verified
//
#include <hip/hip_runtime.h>
#include <stdint.h>

// ---------------------------------------------------------------------------
// Pointer-generator network head for MI455X (gfx1250, wave32, WMMA + async).
//
// out[t,b,v] = p_gen[t,b] * (h[t,b,:]·W[:,v] + bias[v])
//            + (1 - p_gen[t,b]) * scatter_s( att[t,b,s] -> v = post[s,b] )
//
// Memory-bound (~475 MB vs 201 GFLOP fp32) -> fp32 WMMA (16x16x4) GEMM.
// Global->LDS staging uses GLOBAL_LOAD_ASYNC_TO_LDS_B128 (ASYNCcnt) with
// double-buffered LDS: chunk i+1 streams into the alternate buffer while
// chunk i is consumed by v_wmma. Grid order (M fastest) keeps each 128-wide
// W band L2-resident across all 16 M-blocks, so W streams from HBM ~once.
// ---------------------------------------------------------------------------

typedef float v2f __attribute__((ext_vector_type(2)));
typedef float v8f __attribute__((ext_vector_type(8)));

#define T_DIM 64
#define B_DIM 32
#define S_DIM 400
#define V_DIM 32000
#define D_DIM 1536
#define M_DIM (T_DIM * B_DIM)   // 2048 GEMM rows

#define BM 128
#define BN 128
#define KC 16                   // K chunk (4 wmma k-steps of 4)
#define SA 20                   // A tile LDS stride: 80B rows -> 16B-aligned b128
                                // dests, conflict-free float2 fragment reads
#define SB 132                  // B tile LDS stride: 528B rows -> 16B-aligned b128
                                // dests, lane-consecutive fragment reads
#define ABUF (BM * SA)          // 2560 floats
#define BBUF (KC * SB)          // 2112 floats
#define STG  (ABUF + BBUF)      // one pipeline stage (18,688 B)
#define NCHUNK (D_DIM / KC)     // 96

__global__ __launch_bounds__(256)
void pgnet_gemm_wmma(const float* __restrict__ h,     // (M, D) row-major
                     const float* __restrict__ W,     // (D, V) row-major
                     const float* __restrict__ bias,  // (V)
                     const float* __restrict__ pgen,  // (M)
                     float* __restrict__ out)         // (M, V)
{
    __shared__ float lds[2 * STG];   // double-buffered A+B tiles (37,376 B)

    const int tid   = threadIdx.x;
    const int lane  = tid & 31;
    const int wave  = tid >> 5;       // 8 waves
    const int wm    = wave & 1;       // 2 waves across M (64 rows each)
    const int wn    = wave >> 1;      // 4 waves across N (32 cols each)
    const int l16   = lane & 15;
    const int lhalf = lane >> 4;

    const int mb = blockIdx.x * BM;   // M fastest -> W band reused in L2
    const int nb = blockIdx.y * BN;

    v8f acc[4][2];
#pragma unroll
    for (int mi = 0; mi < 4; ++mi)
#pragma unroll
        for (int ni = 0; ni < 2; ++ni)
#pragma unroll
            for (int r = 0; r < 8; ++r)
                acc[mi][ni][r] = 0.0f;

    // 256 threads x 4 async b128 gathers cover the 128x16 A and 16x128 B tiles.
    const int a_row = tid >> 2;            // 0..63 (+64 for second)
    const int a_col = (tid & 3) << 2;
    const int b_k   = tid >> 5;            // 0..7 (+8 for second)
    const int b_c   = (tid & 31) << 2;

    // LDS byte offsets for this lane's async destinations (generic shared
    // pointer truncated to addr[31:0] == LDS byte offset, ISA 10.2).
    const uint32_t ldsBase = (uint32_t)(uintptr_t)lds;
    uint32_t la0[2], la1[2], lb0[2], lb1[2];
#pragma unroll
    for (int p = 0; p < 2; ++p) {
        la0[p] = ldsBase + 4u * (uint32_t)(p * STG + a_row * SA + a_col);
        la1[p] = ldsBase + 4u * (uint32_t)(p * STG + (a_row + 64) * SA + a_col);
        lb0[p] = ldsBase + 4u * (uint32_t)(p * STG + ABUF + b_k * SB + b_c);
        lb1[p] = ldsBase + 4u * (uint32_t)(p * STG + ABUF + (b_k + 8) * SB + b_c);
    }

    // Global sources; advance KC floats (A) / KC rows (B) per chunk.
    const float* gA0 = h + (size_t)(mb + a_row)      * D_DIM + a_col;
    const float* gA1 = h + (size_t)(mb + a_row + 64) * D_DIM + a_col;
    const float* gB0 = W + (size_t)b_k       * V_DIM + nb + b_c;
    const float* gB1 = W + (size_t)(b_k + 8) * V_DIM + nb + b_c;

    // Issue chunk c's four async 16B transfers into LDS stage p.
    auto issue = [&](int c, int p) {
        const uint64_t pa0 = (uint64_t)(uintptr_t)(gA0 + (size_t)c * KC);
        const uint64_t pa1 = (uint64_t)(uintptr_t)(gA1 + (size_t)c * KC);
        const uint64_t pb0 = (uint64_t)(uintptr_t)(gB0 + (size_t)c * KC * V_DIM);
        const uint64_t pb1 = (uint64_t)(uintptr_t)(gB1 + (size_t)c * KC * V_DIM);
        asm volatile("global_load_async_to_lds_b128 %0, %1, off"
                     :: "v"(la0[p]), "v"(pa0) : "memory");
        asm volatile("global_load_async_to_lds_b128 %0, %1, off"
                     :: "v"(la1[p]), "v"(pa1) : "memory");
        asm volatile("global_load_async_to_lds_b128 %0, %1, off"
                     :: "v"(lb0[p]), "v"(pb0) : "memory");
        asm volatile("global_load_async_to_lds_b128 %0, %1, off"
                     :: "v"(lb1[p]), "v"(pb1) : "memory");
    };

    issue(0, 0);   // prologue: stage chunk 0

    for (int i = 0; i < NCHUNK; ++i) {
        const int p = i & 1;

        if (i + 1 < NCHUNK) {
            issue(i + 1, p ^ 1);
            // Async loads complete in order: <=4 outstanding means chunk i's
            // four transfers (issued last iteration) have landed in LDS.
            asm volatile("s_wait_asynccnt 0x4" ::: "memory");
        } else {
            asm volatile("s_wait_asynccnt 0x0" ::: "memory");
        }
        __syncthreads();   // all waves' chunk-i data visible

        const float* bufA = lds + p * STG;
        const float* bufB = bufA + ABUF;

#pragma unroll
        for (int ko = 0; ko < 4; ++ko) {
            // A 16x4 f32 fragment: lane<16 -> K = 4ko+{0,1}; lane>=16 -> +{2,3}
            const int kk = 4 * ko + 2 * lhalf;

            v2f afrag[4];
#pragma unroll
            for (int mi = 0; mi < 4; ++mi) {
                const int row = wm * 64 + mi * 16 + l16;
                afrag[mi] = *(const v2f*)(bufA + row * SA + kk);
            }

            v2f bfrag[2];
#pragma unroll
            for (int ni = 0; ni < 2; ++ni) {
                const int col = wn * 32 + ni * 16 + l16;
                bfrag[ni].x = bufB[kk       * SB + col];
                bfrag[ni].y = bufB[(kk + 1) * SB + col];
            }

#pragma unroll
            for (int mi = 0; mi < 4; ++mi)
#pragma unroll
                for (int ni = 0; ni < 2; ++ni)
                    acc[mi][ni] = __builtin_amdgcn_wmma_f32_16x16x4_f32(
                        /*neg_a=*/false, afrag[mi],
                        /*neg_b=*/false, bfrag[ni],
                        /*c_mod=*/(short)0, acc[mi][ni],
                        /*reuse_a=*/false, /*reuse_b=*/false);
        }

        __syncthreads();   // chunk-i buffer free for the i+2 async writes
    }

    // --- epilogue: out = p_gen * (acc + bias), exactly covers d_out ---
    // C/D layout: VGPR r holds M = r + 8*lhalf, N = lane%16.
#pragma unroll
    for (int ni = 0; ni < 2; ++ni) {
        const int col_g = nb + wn * 32 + ni * 16 + l16;
        const float bv = bias[col_g];
#pragma unroll
        for (int mi = 0; mi < 4; ++mi) {
            const int rowbase = mb + wm * 64 + mi * 16 + 8 * lhalf;
#pragma unroll
            for (int r = 0; r < 8; ++r) {
                const int row_g = rowbase + r;
                const float pg = pgen[row_g];
                out[(size_t)row_g * V_DIM + col_g] = pg * (acc[mi][ni][r] + bv);
            }
        }
    }
}

// Scatter-add of the copy distribution: out[t,b,post[s,b]] += (1-p_gen)*att[t,b,s]
__global__ __launch_bounds__(256)
void pgnet_scatter(const float* __restrict__ att,   // (T,B,S)
                   const float* __restrict__ pgen,  // (T*B)
                   const int*   __restrict__ post,  // (S,B)
                   float* __restrict__ out)         // (T*B, V)
{
    const int tb = blockIdx.x;            // t*B + b
    const int b  = tb & (B_DIM - 1);
    const float scale = 1.0f - pgen[tb];
    float* orow = out + (size_t)tb * V_DIM;
    const float* arow = att + (size_t)tb * S_DIM;

    for (int s = threadIdx.x; s < S_DIM; s += blockDim.x) {
        const int v = post[s * B_DIM + b];
        atomicAdd(orow + v, scale * arow[s]);   // global_atomic_add_f32
    }
}

extern "C" void kernel_launch(void* const* d_in, const int* in_sizes, int n_in,
                              void* d_out, int out_size, void* d_ws, size_t ws_size,
                              hipStream_t stream) {
    const float* h    = (const float*)d_in[0];   // (T,B,D)
    const float* W    = (const float*)d_in[1];   // (D,V)
    const float* bias = (const float*)d_in[2];   // (V)
    const float* att  = (const float*)d_in[3];   // (T,B,S)
    const float* pgen = (const float*)d_in[4];   // (T,B,1)
    const int*   post = (const int*)d_in[5];     // (S,B)
    float* out = (float*)d_out;                  // (T,B,V)

    dim3 grid(M_DIM / BM, V_DIM / BN);           // (16, 250), M fastest
    pgnet_gemm_wmma<<<grid, 256, 0, stream>>>(h, W, bias, pgen, out);
    pgnet_scatter<<<M_DIM, 256, 0, stream>>>(att, pgen, post, out);
}